// GAT_16922171146361
// MI455X (gfx1250) — compile-verified
//
#include <hip/hip_runtime.h>
#include <hip/hip_bf16.h>
#include <math.h>

typedef __attribute__((ext_vector_type(2))) float v2f;
typedef __attribute__((ext_vector_type(8))) float v8f;

#define NEG_SLOPE 0.2f
#define EPSV 1e-16f

// ---------------------------------------------------------------- utilities

__global__ void fill_kernel(float* __restrict__ p, float v, int n) {
  int i = blockIdx.x * blockDim.x + threadIdx.x;
  if (i < n) p[i] = v;
}

__device__ __forceinline__ void atomicMaxF(float* addr, float val) {
  // ordered-int trick: works with init = -inf
  if (val >= 0.0f) atomicMax((int*)addr, __float_as_int(val));
  else             atomicMin((unsigned int*)addr, __float_as_uint(val));
}

__device__ __forceinline__ void edge_sd(const int* __restrict__ ei, int E, int idx,
                                        int& s, int& d) {
  if (idx < E) { s = ei[idx]; d = ei[E + idx]; }
  else         { s = idx - E; d = idx - E; }   // appended self loops
}

// ------------------------------------------------------- WMMA f32 GEMM tile
// One wave computes a 16 x (16*NT) output slab: A is loaded ONCE per K-step
// and reused across NT accumulators (4x less traffic on the big A operand).
// D[M x ldd] (valid cols < nvalid) = A[M x K] (row major) * B[K x ldb].
// Uses V_WMMA_F32_16X16X4_F32 (native fp32 matrix op, wave32).
template <int NT>
__global__ void __launch_bounds__(32)
gemm_wmma_f32_nt(const float* __restrict__ A, const float* __restrict__ B,
                 float* __restrict__ D, int M, int K, int ldb, int ldd, int nvalid) {
  const int tileM = blockIdx.x;
  const int lane = threadIdx.x & 31;
  const int half = lane >> 4;      // 0: K pair {0,1}, 1: K pair {2,3}
  const int l16  = lane & 15;
  const int rowA = tileM * 16 + l16;
  const float* __restrict__ arow = A + (size_t)rowA * K;

  v8f acc[NT];
  #pragma unroll
  for (int nt = 0; nt < NT; ++nt) acc[nt] = (v8f){0.f,0.f,0.f,0.f,0.f,0.f,0.f,0.f};

  for (int kk = 0; kk < K; kk += 4) {
    const int ka = kk + half * 2;
    v2f a;
    a.x = arow[ka];
    a.y = arow[ka + 1];
    const float* __restrict__ brow0 = B + (size_t)ka * ldb;
    const float* __restrict__ brow1 = B + (size_t)(ka + 1) * ldb;
    #pragma unroll
    for (int nt = 0; nt < NT; ++nt) {
      const int colB = nt * 16 + l16;
      v2f b;
      b.x = brow0[colB];
      b.y = brow1[colB];
      acc[nt] = __builtin_amdgcn_wmma_f32_16x16x4_f32(false, a, false, b,
                                                      (short)0, acc[nt], false, false);
    }
  }
  // C/D layout: VGPR g -> row = g + 8*half, col = l16 (per ISA 7.12.2)
  #pragma unroll
  for (int nt = 0; nt < NT; ++nt) {
    #pragma unroll
    for (int g = 0; g < 8; ++g) {
      const int r = tileM * 16 + half * 8 + g;
      const int c = nt * 16 + l16;
      if (c < nvalid) D[(size_t)r * ldd + c] = acc[nt][g];
    }
  }
}

// zero-pad W2 (K x Nin) -> (K x Nout)
__global__ void pad_kernel(const float* __restrict__ W, float* __restrict__ Wp,
                           int K, int Nin, int Nout) {
  int i = blockIdx.x * blockDim.x + threadIdx.x;
  if (i >= K * Nout) return;
  int k = i / Nout, c = i % Nout;
  Wp[i] = (c < Nin) ? W[k * Nin + c] : 0.0f;
}

// -------------------------------------------------------------- layer 1 ops

// per-(node,head) attention logits: as[n,h] = <h1[n,h,:], a_src[h,:]>
__global__ void alphas1_kernel(const float* __restrict__ h,
                               const float* __restrict__ a_src,
                               const float* __restrict__ a_dst,
                               float* __restrict__ as, float* __restrict__ ad, int N) {
  int i = blockIdx.x * blockDim.x + threadIdx.x;   // i = n*8 + head
  if (i >= N * 8) return;
  int n = i >> 3, hh = i & 7;
  const float* hp = h + (size_t)n * 64 + hh * 8;
  const float* sp = a_src + hh * 8;
  const float* dp = a_dst + hh * 8;
  float s1 = 0.f, s2 = 0.f;
  #pragma unroll
  for (int c = 0; c < 8; ++c) { float v = hp[c]; s1 += v * sp[c]; s2 += v * dp[c]; }
  as[i] = s1; ad[i] = s2;
}

__global__ void edge_max1_kernel(const int* __restrict__ ei, int E, int Et,
                                 const float* __restrict__ as, const float* __restrict__ ad,
                                 float* __restrict__ m) {
  int idx = blockIdx.x * blockDim.x + threadIdx.x;
  if (idx >= Et) return;
  int s, d; edge_sd(ei, E, idx, s, d);
  #pragma unroll
  for (int h = 0; h < 8; ++h) {
    float v = as[s * 8 + h] + ad[d * 8 + h];
    v = v > 0.f ? v : NEG_SLOPE * v;
    atomicMaxF(&m[d * 8 + h], v);
  }
}

__global__ void edge_sum1_kernel(const int* __restrict__ ei, int E, int Et,
                                 const float* __restrict__ as, const float* __restrict__ ad,
                                 const float* __restrict__ m, float* __restrict__ den) {
  int idx = blockIdx.x * blockDim.x + threadIdx.x;
  if (idx >= Et) return;
  int s, d; edge_sd(ei, E, idx, s, d);
  #pragma unroll
  for (int h = 0; h < 8; ++h) {
    float v = as[s * 8 + h] + ad[d * 8 + h];
    v = v > 0.f ? v : NEG_SLOPE * v;
    atomicAdd(&den[d * 8 + h], expf(v - m[d * 8 + h]));
  }
}

__global__ void edge_agg1_kernel(const int* __restrict__ ei, int E, int Et,
                                 const float* __restrict__ as, const float* __restrict__ ad,
                                 const float* __restrict__ m, const float* __restrict__ den,
                                 const float* __restrict__ h1, float* __restrict__ out) {
  int t = blockIdx.x * blockDim.x + threadIdx.x;   // t = edge*8 + head
  if (t >= Et * 8) return;
  int idx = t >> 3, hh = t & 7;
  int s, d; edge_sd(ei, E, idx, s, d);
  float v = as[s * 8 + hh] + ad[d * 8 + hh];
  v = v > 0.f ? v : NEG_SLOPE * v;
  float alpha = expf(v - m[d * 8 + hh]) / (den[d * 8 + hh] + EPSV);
  const float* hp = h1 + (size_t)s * 64 + hh * 8;
  float* op = out + (size_t)d * 64 + hh * 8;
  #pragma unroll
  for (int c = 0; c < 8; ++c) atomicAdd(&op[c], hp[c] * alpha);
}

__global__ void bias_elu_kernel(float* __restrict__ x, const float* __restrict__ b,
                                int total, int C) {
  int i = blockIdx.x * blockDim.x + threadIdx.x;
  if (i >= total) return;
  float v = x[i] + b[i % C];
  x[i] = v > 0.f ? v : expm1f(v);
}

// -------------------------------------------------------------- layer 2 ops

__global__ void alphas2_kernel(const float* __restrict__ h,
                               const float* __restrict__ a_src,
                               const float* __restrict__ a_dst,
                               float* __restrict__ as, float* __restrict__ ad,
                               int N, int C) {
  int n = blockIdx.x * blockDim.x + threadIdx.x;
  if (n >= N) return;
  float s1 = 0.f, s2 = 0.f;
  for (int c = 0; c < C; ++c) { float v = h[(size_t)n * C + c]; s1 += v * a_src[c]; s2 += v * a_dst[c]; }
  as[n] = s1; ad[n] = s2;
}

__global__ void edge_max2_kernel(const int* __restrict__ ei, int E, int Et,
                                 const float* __restrict__ as, const float* __restrict__ ad,
                                 float* __restrict__ m) {
  int idx = blockIdx.x * blockDim.x + threadIdx.x;
  if (idx >= Et) return;
  int s, d; edge_sd(ei, E, idx, s, d);
  float v = as[s] + ad[d];
  v = v > 0.f ? v : NEG_SLOPE * v;
  atomicMaxF(&m[d], v);
}

__global__ void edge_sum2_kernel(const int* __restrict__ ei, int E, int Et,
                                 const float* __restrict__ as, const float* __restrict__ ad,
                                 const float* __restrict__ m, float* __restrict__ den) {
  int idx = blockIdx.x * blockDim.x + threadIdx.x;
  if (idx >= Et) return;
  int s, d; edge_sd(ei, E, idx, s, d);
  float v = as[s] + ad[d];
  v = v > 0.f ? v : NEG_SLOPE * v;
  atomicAdd(&den[d], expf(v - m[d]));
}

__global__ void edge_agg2_kernel(const int* __restrict__ ei, int E, int Et,
                                 const float* __restrict__ as, const float* __restrict__ ad,
                                 const float* __restrict__ m, const float* __restrict__ den,
                                 const float* __restrict__ h2, float* __restrict__ out, int C) {
  int idx = blockIdx.x * blockDim.x + threadIdx.x;
  if (idx >= Et) return;
  int s, d; edge_sd(ei, E, idx, s, d);
  float v = as[s] + ad[d];
  v = v > 0.f ? v : NEG_SLOPE * v;
  float alpha = expf(v - m[d]) / (den[d] + EPSV);
  const float* hp = h2 + (size_t)s * C;
  float* op = out + (size_t)d * C;
  for (int c = 0; c < C; ++c) atomicAdd(&op[c], hp[c] * alpha);
}

// bias + elu + log_softmax over the class dim
__global__ void final_kernel(const float* __restrict__ out2, const float* __restrict__ b2,
                             float* __restrict__ y, int N, int C) {
  int n = blockIdx.x * blockDim.x + threadIdx.x;
  if (n >= N) return;
  float buf[40];
  float mx = -INFINITY;
  for (int c = 0; c < C; ++c) {
    float v = out2[(size_t)n * C + c] + b2[c];
    v = v > 0.f ? v : expm1f(v);
    buf[c] = v;
    mx = fmaxf(mx, v);
  }
  float sm = 0.f;
  for (int c = 0; c < C; ++c) sm += expf(buf[c] - mx);
  float l = logf(sm);
  for (int c = 0; c < C; ++c) y[(size_t)n * C + c] = buf[c] - mx - l;
}

// ------------------------------------------------------------------- driver

static inline int cdiv(int a, int b) { return (a + b - 1) / b; }

extern "C" void kernel_launch(void* const* d_in, const int* in_sizes, int n_in,
                              void* d_out, int out_size, void* d_ws, size_t ws_size,
                              hipStream_t stream) {
  const float* x      = (const float*)d_in[0];
  const int*   ei     = (const int*)  d_in[1];
  const float* W1     = (const float*)d_in[2];
  const float* a_src1 = (const float*)d_in[3];
  const float* a_dst1 = (const float*)d_in[4];
  const float* b1     = (const float*)d_in[5];
  const float* W2     = (const float*)d_in[6];
  const float* a_src2 = (const float*)d_in[7];
  const float* a_dst2 = (const float*)d_in[8];
  const float* b2     = (const float*)d_in[9];
  float* out = (float*)d_out;

  const int HC   = in_sizes[5];              // 64  (= heads * out_ch)
  const int NCLS = in_sizes[9];              // 40
  const int F    = in_sizes[2] / HC;         // 256
  const int N    = in_sizes[0] / F;          // 50000
  const int E    = in_sizes[1] / 2;          // 800000
  const int Et   = E + N;                    // + self loops
  const int NPAD = 48;                       // NCLS padded to WMMA tile multiple

  // workspace layout (floats)
  float* ws = (float*)d_ws;
  size_t off = 0;
  float* h1   = ws + off; off += (size_t)N * HC;
  float* as1  = ws + off; off += (size_t)N * 8;
  float* ad1  = ws + off; off += (size_t)N * 8;
  float* m1   = ws + off; off += (size_t)N * 8;
  float* den1 = ws + off; off += (size_t)N * 8;
  float* out1 = ws + off; off += (size_t)N * HC;
  float* h2   = ws + off; off += (size_t)N * NCLS;
  float* as2  = ws + off; off += (size_t)N;
  float* ad2  = ws + off; off += (size_t)N;
  float* m2   = ws + off; off += (size_t)N;
  float* den2 = ws + off; off += (size_t)N;
  float* out2 = ws + off; off += (size_t)N * NCLS;
  float* Wp   = ws + off; off += (size_t)HC * NPAD;

  const int B = 256;

  // init accumulators
  fill_kernel<<<cdiv(N * 8, B), B, 0, stream>>>(m1, -INFINITY, N * 8);
  fill_kernel<<<cdiv(N * 8, B), B, 0, stream>>>(den1, 0.f, N * 8);
  fill_kernel<<<cdiv(N * HC, B), B, 0, stream>>>(out1, 0.f, N * HC);
  fill_kernel<<<cdiv(N, B), B, 0, stream>>>(m2, -INFINITY, N);
  fill_kernel<<<cdiv(N, B), B, 0, stream>>>(den2, 0.f, N);
  fill_kernel<<<cdiv(N * NCLS, B), B, 0, stream>>>(out2, 0.f, N * NCLS);

  // ---- layer 1 ----
  // h1 = x @ W1   (50000x256 @ 256x64): one wave per 16x64 slab, 4 accumulators
  gemm_wmma_f32_nt<4><<<dim3(N / 16), 32, 0, stream>>>(x, W1, h1, N, F, HC, HC, HC);
  alphas1_kernel<<<cdiv(N * 8, B), B, 0, stream>>>(h1, a_src1, a_dst1, as1, ad1, N);
  edge_max1_kernel<<<cdiv(Et, B), B, 0, stream>>>(ei, E, Et, as1, ad1, m1);
  edge_sum1_kernel<<<cdiv(Et, B), B, 0, stream>>>(ei, E, Et, as1, ad1, m1, den1);
  edge_agg1_kernel<<<cdiv(Et * 8, B), B, 0, stream>>>(ei, E, Et, as1, ad1, m1, den1, h1, out1);
  bias_elu_kernel<<<cdiv(N * HC, B), B, 0, stream>>>(out1, b1, N * HC, HC);

  // ---- layer 2 ----
  // pad W2 to 64x48 so the WMMA inner loop has no divergence, then h2 = out1 @ W2
  pad_kernel<<<cdiv(HC * NPAD, B), B, 0, stream>>>(W2, Wp, HC, NCLS, NPAD);
  gemm_wmma_f32_nt<3><<<dim3(N / 16), 32, 0, stream>>>(out1, Wp, h2, N, HC, NPAD, NCLS, NCLS);
  alphas2_kernel<<<cdiv(N, B), B, 0, stream>>>(h2, a_src2, a_dst2, as2, ad2, N, NCLS);
  edge_max2_kernel<<<cdiv(Et, B), B, 0, stream>>>(ei, E, Et, as2, ad2, m2);
  edge_sum2_kernel<<<cdiv(Et, B), B, 0, stream>>>(ei, E, Et, as2, ad2, m2, den2);
  edge_agg2_kernel<<<cdiv(Et, B), B, 0, stream>>>(ei, E, Et, as2, ad2, m2, den2, h2, out2, NCLS);

  final_kernel<<<cdiv(N, B), B, 0, stream>>>(out2, b2, out, N, NCLS);
}